// VQ_28733331210717
// MI455X (gfx1250) — compile-verified
//
#include <hip/hip_runtime.h>

typedef float v2f __attribute__((ext_vector_type(2)));
typedef float v8f __attribute__((ext_vector_type(8)));

#define B_    8
#define T_    4096
#define C_    64
#define V_    8192
#define NTOK  (B_ * T_)
#define LDSROW 68   // 64 + 4 pad floats -> conflict-free ds_load_b64 B-fragment reads

// ---------- kernel 1: codebook squared norms ----------
__global__ void vq_c2_kernel(const float* __restrict__ cb, float* __restrict__ c2) {
    int v = blockIdx.x * blockDim.x + threadIdx.x;
    if (v >= V_) return;
    const float4* row = (const float4*)(cb + (size_t)v * C_);
    float s = 0.f;
#pragma unroll
    for (int i = 0; i < C_ / 4; ++i) {
        float4 q = row[i];
        s += q.x * q.x + q.y * q.y + q.z * q.z + q.w * q.w;
    }
    c2[v] = s;
}

// ---------- kernel 2: fused GEMM (fp32 WMMA) + rowwise argmin ----------
// Each wave: one 16-token row tile (A in registers). Workgroup (8 waves = 128 tokens)
// stages 128-code x 64-dim codebook tiles through LDS (padded stride).
__launch_bounds__(256)
__global__ void vq_argmin_kernel(const float* __restrict__ x,
                                 const float* __restrict__ cb,
                                 const float* __restrict__ c2,
                                 int* __restrict__ idx_out) {
    __shared__ float lds[128 * LDSROW];   // ~34 KB of the 320 KB WGP LDS

    const int lane  = threadIdx.x & 31;
    const int wave  = threadIdx.x >> 5;
    const int nsub  = lane & 15;   // M (for A) / N (for B) within tile
    const int khalf = lane >> 4;   // K pair selector (0 -> K,K+1 ; 1 -> K+2,K+3)
    const int t0    = blockIdx.x * 128 + wave * 16;

    // A fragments: 16 K-steps of float2 (32 VGPRs), per ISA 32-bit A 16x4 layout.
    v2f a[16];
    const float* xrow = x + (size_t)(t0 + nsub) * C_ + khalf * 2;
#pragma unroll
    for (int ks = 0; ks < 16; ++ks)
        a[ks] = *(const v2f*)(xrow + ks * 4);

    float bestv[8];
    int   besti[8];
#pragma unroll
    for (int r = 0; r < 8; ++r) { bestv[r] = 3.4028235e38f; besti[r] = 0; }

    const int row2 = threadIdx.x >> 1;      // staging: 2 threads per codebook row
    const int half = threadIdx.x & 1;

    for (int chunk = 0; chunk < V_ / 128; ++chunk) {
        const float* src = cb + (size_t)chunk * 128 * C_;
        __syncthreads();                    // protect LDS reads of previous chunk
        // stage 128x64 f32 -> padded LDS: each thread copies half a row (8 float4)
#pragma unroll
        for (int j = 0; j < 8; ++j) {
            float4 q = ((const float4*)src)[row2 * 16 + half * 8 + j];
            *(float4*)(lds + row2 * LDSROW + half * 32 + j * 4) = q;
        }
        __syncthreads();

        if (chunk + 1 < V_ / 128)           // global_prefetch_b8 for next tile
            __builtin_prefetch(cb + (size_t)(chunk + 1) * 128 * C_ + threadIdx.x * 32, 0, 1);

#pragma unroll
        for (int sub = 0; sub < 8; ++sub) {
            const int v0 = chunk * 128 + sub * 16;
            const float* brow = lds + (sub * 16 + nsub) * LDSROW + khalf * 2;
            v8f acc = {0.f, 0.f, 0.f, 0.f, 0.f, 0.f, 0.f, 0.f};
#pragma unroll
            for (int ks = 0; ks < 16; ++ks) {
                v2f b = *(const v2f*)(brow + ks * 4);
                acc = __builtin_amdgcn_wmma_f32_16x16x4_f32(
                        false, a[ks], false, b, (short)0, acc, false, false);
            }
            const float cc  = c2[v0 + nsub];
            const int   col = v0 + nsub;
#pragma unroll
            for (int r = 0; r < 8; ++r) {
                float s = __builtin_fmaf(-2.f, acc[r], cc);   // c2 - 2*dot
                if (s < bestv[r]) { bestv[r] = s; besti[r] = col; }  // first-min kept
            }
        }
    }

    // cross-lane argmin: xor masks 1..8 stay inside each 16-lane half (wave32)
#pragma unroll
    for (int off = 1; off < 16; off <<= 1) {
#pragma unroll
        for (int r = 0; r < 8; ++r) {
            float ov = __shfl_xor(bestv[r], off, 32);
            int   oi = __shfl_xor(besti[r], off, 32);
            if (ov < bestv[r] || (ov == bestv[r] && oi < besti[r])) {
                bestv[r] = ov; besti[r] = oi;
            }
        }
    }
    if (nsub == 0) {
        const int mbase = khalf * 8;   // lane0 -> rows 0..7, lane16 -> rows 8..15
#pragma unroll
        for (int r = 0; r < 8; ++r)
            idx_out[t0 + mbase + r] = besti[r];
    }
}

// ---------- kernel 3: gather + STE output + per-token squared error ----------
__launch_bounds__(256)
__global__ void vq_gather_kernel(const float* __restrict__ x,
                                 const float* __restrict__ cb,
                                 const int* __restrict__ idx_ws,
                                 float* __restrict__ out_ste,
                                 float* __restrict__ out_idx,
                                 float* __restrict__ tokd) {
    const int lane  = threadIdx.x & 31;
    const int token = blockIdx.x * 8 + (threadIdx.x >> 5);
    const int vi    = idx_ws[token];

    const float2 e  = *(const float2*)(cb + (size_t)vi * C_ + lane * 2);
    const float2 xv = *(const float2*)(x + (size_t)token * C_ + lane * 2);
    *(float2*)(out_ste + (size_t)token * C_ + lane * 2) = e;   // vq_ste == vq_embed

    float dx = e.x - xv.x, dy = e.y - xv.y;
    float d  = dx * dx + dy * dy;
#pragma unroll
    for (int off = 16; off; off >>= 1) d += __shfl_xor(d, off, 32);

    if (lane == 0) {
        out_idx[token] = (float)vi;   // exact for vi < 2^24
        tokd[token]    = d;
    }
}

// ---------- kernel 4: deterministic per-batch loss reduction ----------
__global__ void vq_loss_kernel(const float* __restrict__ tokd,
                               float* __restrict__ out_loss) {
    __shared__ float red[256];
    const int b = blockIdx.x;            // batch 0..7
    float s = 0.f;
    for (int i = threadIdx.x; i < T_; i += 256) s += tokd[b * T_ + i];
    red[threadIdx.x] = s;
    __syncthreads();
    for (int st = 128; st; st >>= 1) {
        if (threadIdx.x < st) red[threadIdx.x] += red[threadIdx.x + st];
        __syncthreads();
    }
    if (threadIdx.x == 0) {
        float m = red[0] / (float)(T_ * C_);
        out_loss[b]     = m;   // vq_loss
        out_loss[8 + b] = m;   // commit_loss (identical forward value)
    }
}

extern "C" void kernel_launch(void* const* d_in, const int* in_sizes, int n_in,
                              void* d_out, int out_size, void* d_ws, size_t ws_size,
                              hipStream_t stream) {
    (void)in_sizes; (void)n_in; (void)out_size; (void)ws_size;
    const float* x  = (const float*)d_in[0];
    const float* cb = (const float*)d_in[1];

    float* out      = (float*)d_out;
    float* out_ste  = out;                              // [B,T,C]
    float* out_idx  = out + (size_t)NTOK * C_;          // [B,T] (as float)
    float* out_loss = out_idx + NTOK;                   // [8] vq + [8] commit

    int*   idx_ws = (int*)d_ws;                         // 32768 ints
    float* c2     = (float*)d_ws + NTOK;                // 8192 floats
    float* tokd   = c2 + V_;                            // 32768 floats (~288 KB total)

    vq_c2_kernel    <<<V_ / 256,   256, 0, stream>>>(cb, c2);
    vq_argmin_kernel<<<NTOK / 128, 256, 0, stream>>>(x, cb, c2, idx_ws);
    vq_gather_kernel<<<NTOK / 8,   256, 0, stream>>>(x, cb, idx_ws, out_ste, out_idx, tokd);
    vq_loss_kernel  <<<8,          256, 0, stream>>>(tokd, out_loss);
}